// ContrastiveLoss_78030965834244
// MI455X (gfx1250) — compile-verified
//
#include <hip/hip_runtime.h>
#include <math.h>

typedef float v2f __attribute__((ext_vector_type(2)));
typedef float v8f __attribute__((ext_vector_type(8)));

#define NB 32       // splits
#define NL 1024     // candidates per split
#define ND 256      // embed dim
#define NP 8        // positives per split
#define MARGIN_F 0.01f

// ---------------------------------------------------------------------------
// Kernel 1: L2-normalize the queries.  One block (256 thr) per split.
// ---------------------------------------------------------------------------
__global__ void cl_qnorm_kernel(const float* __restrict__ q, float* __restrict__ qn) {
    const int b    = blockIdx.x;
    const int tid  = threadIdx.x;          // 0..255 == d
    float v  = q[b * ND + tid];
    float ss = v * v;
    #pragma unroll
    for (int off = 16; off > 0; off >>= 1) ss += __shfl_xor(ss, off, 32);
    __shared__ float ws[8];
    if ((tid & 31) == 0) ws[tid >> 5] = ss;
    __syncthreads();
    float tot = 0.f;
    #pragma unroll
    for (int i = 0; i < 8; ++i) tot += ws[i];
    float n = fmaxf(sqrtf(tot), 1e-12f);
    qn[b * ND + tid] = v / n;
}

// ---------------------------------------------------------------------------
// Kernel 2: cosine sims via V_WMMA_F32_16X16X4_F32.
// grid = (NL/128, NB), block = 256 (8 waves).  Wave w handles a 16-row tile.
// A operand: lane m   (0..15) holds E[row=m, k+0..1]
//            lane m+16       holds E[row=m, k+2..3]   (koff = (lane>>4)*2)
// B operand: same addressing into the normalized query -> every D column is
//            dot(row, q); extract from D (lane 0: M=0..7, lane 16: M=8..15).
// ---------------------------------------------------------------------------
__global__ void cl_sim_wmma_kernel(const float* __restrict__ E,
                                   const float* __restrict__ qn,
                                   float* __restrict__ sim) {
    __shared__ __align__(16) float qs[ND];
    __shared__ float xdot[8][16];
    __shared__ float xssq[8][16];

    const int b     = blockIdx.y;
    const int wave  = threadIdx.x >> 5;
    const int lane  = threadIdx.x & 31;
    const int tile  = blockIdx.x * 8 + wave;        // 16-row tile index (0..63)
    const int rloc  = lane & 15;
    const int row   = tile * 16 + rloc;
    const int koff  = (lane >> 4) * 2;

    // stage normalized query for this split into LDS (uniform, no divergence)
    for (int i = threadIdx.x; i < ND; i += blockDim.x) qs[i] = qn[b * ND + i];
    __syncthreads();

    const float* erow = E + ((size_t)b * NL + row) * ND + koff;
    const float* qrow = qs + koff;

    v8f  acc = {};            // C/D accumulator, starts at 0
    float ssq = 0.f;          // per-lane partial row sum-of-squares

    #pragma unroll 8
    for (int k = 0; k < ND; k += 4) {
        __builtin_prefetch(erow + k + 64, 0, 0);     // global_prefetch ahead
        v2f a  = *(const v2f*)(erow + k);            // global_load_b64
        v2f bq = *(const v2f*)(qrow + k);            // ds_load_b64
        ssq += a.x * a.x + a.y * a.y;
        // D = A(16x4 f32) * B(4x16 f32, q replicated across columns) + C
        acc = __builtin_amdgcn_wmma_f32_16x16x4_f32(
            /*neg_a=*/false, a, /*neg_b=*/false, bq,
            /*c_mod=*/(short)0, acc, /*reuse_a=*/false, /*reuse_b=*/false);
    }

    // fold row sumsq across the lane pair (lane m has k%4 in {0,1}; m+16 has {2,3})
    ssq += __shfl_xor(ssq, 16, 32);
    if (lane < 16) xssq[wave][rloc] = ssq;

    // D layout: VGPR r -> lanes 0..15: M=r ; lanes 16..31: M=r+8 (N = lane%16)
    if (lane == 0) {
        #pragma unroll
        for (int r = 0; r < 8; ++r) xdot[wave][r] = acc[r];
    }
    if (lane == 16) {
        #pragma unroll
        for (int r = 0; r < 8; ++r) xdot[wave][8 + r] = acc[r];
    }
    __syncthreads();

    if (lane < 16) {
        float n = fmaxf(sqrtf(xssq[wave][rloc]), 1e-12f);
        sim[(size_t)b * NL + tile * 16 + rloc] = xdot[wave][rloc] / n;
    }
}

// ---------------------------------------------------------------------------
// Kernel 3: per-split hinge sum.  One block (256 thr) per split.
// loss_b = sum_{p in pos} sum_{n not in pos} relu(sim[n] - sim[p] + margin)
// ---------------------------------------------------------------------------
__global__ void cl_hinge_kernel(const float* __restrict__ sim,
                                const int*  __restrict__ pos_idx,
                                float* __restrict__ partials) {
    const int b   = blockIdx.x;
    const int tid = threadIdx.x;
    __shared__ float psim[NP];
    __shared__ int   pidx[NP];
    if (tid < NP) {
        int idx  = pos_idx[b * NP + tid];
        pidx[tid] = idx;
        psim[tid] = sim[(size_t)b * NL + idx];
    }
    __syncthreads();

    float acc = 0.f;
    for (int l = tid; l < NL; l += blockDim.x) {
        bool is_pos = false;
        #pragma unroll
        for (int p = 0; p < NP; ++p) is_pos |= (pidx[p] == l);
        if (!is_pos) {
            float s = sim[(size_t)b * NL + l];
            #pragma unroll
            for (int p = 0; p < NP; ++p)
                acc += fmaxf(s - psim[p] + MARGIN_F, 0.f);
        }
    }
    #pragma unroll
    for (int off = 16; off > 0; off >>= 1) acc += __shfl_xor(acc, off, 32);
    __shared__ float ws[8];
    if ((tid & 31) == 0) ws[tid >> 5] = acc;
    __syncthreads();
    if (tid == 0) {
        float t = 0.f;
        #pragma unroll
        for (int i = 0; i < 8; ++i) t += ws[i];
        partials[b] = t;
    }
}

// ---------------------------------------------------------------------------
// Kernel 4: deterministic final reduce (single wave32), divide by pair count.
// ---------------------------------------------------------------------------
__global__ void cl_final_kernel(const float* __restrict__ partials,
                                float* __restrict__ out) {
    float v = partials[threadIdx.x];   // 32 lanes, 32 splits
    #pragma unroll
    for (int off = 16; off > 0; off >>= 1) v += __shfl_xor(v, off, 32);
    if (threadIdx.x == 0)
        out[0] = v / (float)(NB * NP * (NL - NP));   // 260096 pairs
}

// ---------------------------------------------------------------------------
// Launch
// ---------------------------------------------------------------------------
extern "C" void kernel_launch(void* const* d_in, const int* in_sizes, int n_in,
                              void* d_out, int out_size, void* d_ws, size_t ws_size,
                              hipStream_t stream) {
    const float* E   = (const float*)d_in[0];   // [32,1024,256]
    const float* q   = (const float*)d_in[1];   // [32,256]
    const int*   pos = (const int*)  d_in[2];   // [32,8]
    (void)in_sizes; (void)n_in; (void)out_size; (void)ws_size;

    float* qn       = (float*)d_ws;                       //  32 KB
    float* sim      = qn  + NB * ND;                      // 128 KB
    float* partials = sim + NB * NL;                      // 128 B

    cl_qnorm_kernel   <<<NB, ND, 0, stream>>>(q, qn);
    cl_sim_wmma_kernel<<<dim3(NL / 128, NB), 256, 0, stream>>>(E, qn, sim);
    cl_hinge_kernel   <<<NB, 256, 0, stream>>>(sim, pos, partials);
    cl_final_kernel   <<<1, 32, 0, stream>>>(partials, (float*)d_out);
}